// NCELanguageModelLoss_39273180955639
// MI455X (gfx1250) — compile-verified
//
#include <hip/hip_runtime.h>
#include <hip/hip_bf16.h>
#include <math.h>

typedef __attribute__((ext_vector_type(2))) float v2f;
typedef __attribute__((ext_vector_type(8))) float v8f;

// numerically stable log_sigmoid(x) = min(x,0) - log1p(exp(-|x|))
__device__ __forceinline__ float lsig(float x) {
    return fminf(x, 0.0f) - log1pf(expf(-fabsf(x)));
}

// One wave per 16-row tile of hidden. 8 waves / block -> 128 rows per block.
// LDS holds noise^T padded to 32 columns, layout idx = ((k>>1)*32 + n)*2 + (k&1)
// so each B fragment (2 consecutive k for one column) is one 8-byte LDS load.
__global__ void nce_main_kernel(const float* __restrict__ h,
                                const int*   __restrict__ tgt,
                                const float* __restrict__ W,
                                const int*   __restrict__ noise,
                                float*       __restrict__ partials,
                                int n, int D, int V, int K) {
    extern __shared__ float smem[];          // D*32 floats noise^T + 16 floats reduce
    float* red = smem + (size_t)D * 32;

    const int tid  = threadIdx.x;
    const int wid  = tid >> 5;
    const int lane = tid & 31;
    const int m    = lane & 15;              // row-in-tile (A) / column (B)
    const int kp   = lane >> 4;              // k-pair selector (K=0,1 vs K=2,3)

    // ---- stage noise^T into LDS (zero-pad columns K..31) ----
    for (int i = tid; i < D * 32; i += blockDim.x) {
        const int klo = i & 1;
        const int nn  = (i >> 1) & 31;
        const int kk  = ((i >> 6) << 1) | klo;
        float v = 0.0f;
        if (nn < K) v = W[(size_t)noise[nn] * D + kk];
        smem[i] = v;                         // fully coalesced ds writes
    }
    __syncthreads();

    const int rowBase = blockIdx.x * 128 + wid * 16;

    // fused pointers: this lane's slice of hidden row (m) and its target W row
    const float* __restrict__ hA = h + (size_t)(rowBase + m) * D + 2 * kp;
    const float* __restrict__ wT = W + (size_t)tgt[rowBase + m] * D + 2 * kp;

    v8f acc0 = {};                           // noise cols 0..15
    v8f acc1 = {};                           // noise cols 16..31 (only <K valid)
    float posPart = 0.0f;                    // partial dot(h[row+m], W[tgt]) on this lane

    for (int k0 = 0; k0 < D; k0 += 4) {
        v2f a = *(const v2f*)(hA + k0);                          // global_load_b64
        const float* bp = smem + (size_t)(((k0 >> 1) + kp) * 32) * 2;
        v2f b0 = *(const v2f*)(bp + 2 * m);                      // ds_load_b64
        v2f b1 = *(const v2f*)(bp + 2 * (m + 16));               // ds_load_b64
        v2f wv = *(const v2f*)(wT + k0);                         // global_load_b64

        acc0 = __builtin_amdgcn_wmma_f32_16x16x4_f32(
                   false, a, false, b0, (short)0, acc0, false, false);
        acc1 = __builtin_amdgcn_wmma_f32_16x16x4_f32(
                   false, a, false, b1, (short)0, acc1, false, false);

        posPart = fmaf(a.x, wv.x, posPart);
        posPart = fmaf(a.y, wv.y, posPart);
    }

    // shift C: scores - log(K) - log_unif = scores - (log K - log V)
    const float C = logf((float)K) - logf((float)V);

    // ---- positive terms: lanes m and m+16 together hold full dot of row m ----
    float posFull = posPart + __shfl_down(posPart, 16, 32);
    float posLoc  = (lane < 16) ? lsig(posFull - C) : 0.0f;

    // ---- negative terms from WMMA accumulators ----
    // C/D layout: VGPR r -> row M=r (lanes 0-15) / M=r+8 (lanes 16-31), col = lane%16
    float negLoc = 0.0f;
    #pragma unroll
    for (int r = 0; r < 8; ++r) {
        if (m < K)      negLoc += lsig(-(acc0[r] - C));
        if (16 + m < K) negLoc += lsig(-(acc1[r] - C));
    }

    // ---- wave reduction (wave32) ----
    #pragma unroll
    for (int off = 16; off > 0; off >>= 1) {
        negLoc += __shfl_down(negLoc, off, 32);
        posLoc += __shfl_down(posLoc, off, 32);
    }
    if (lane == 0) { red[wid] = negLoc; red[8 + wid] = posLoc; }
    __syncthreads();

    if (tid == 0) {
        float np = 0.0f, pp = 0.0f;
        #pragma unroll
        for (int w = 0; w < 8; ++w) { np += red[w]; pp += red[8 + w]; }
        partials[(size_t)blockIdx.x * 2 + 0] = pp;
        partials[(size_t)blockIdx.x * 2 + 1] = np;
    }
}

__global__ void nce_final_kernel(const float* __restrict__ partials,
                                 float* __restrict__ out,
                                 int nblocks, int n, int K) {
    if (threadIdx.x == 0 && blockIdx.x == 0) {
        float pp = 0.0f, np = 0.0f;
        for (int i = 0; i < nblocks; ++i) {
            pp += partials[2 * i + 0];
            np += partials[2 * i + 1];
        }
        out[0] = -(pp / (float)n) - (np / ((float)n * (float)K));
    }
}

extern "C" void kernel_launch(void* const* d_in, const int* in_sizes, int n_in,
                              void* d_out, int out_size, void* d_ws, size_t ws_size,
                              hipStream_t stream) {
    const float* hidden = (const float*)d_in[0];   // (B*T, D)
    const int*   tgt    = (const int*)  d_in[1];   // (B*T,)
    const float* W      = (const float*)d_in[2];   // (V, D)
    const int*   noise  = (const int*)  d_in[3];   // (K,)
    float*       out    = (float*)d_out;
    float*       parts  = (float*)d_ws;

    const int n = in_sizes[1];            // 2048
    const int D = in_sizes[0] / n;        // 1024
    const int K = in_sizes[3];            // 20
    const int V = in_sizes[2] / D;        // 50257

    const int rowsPerBlock = 128;         // 8 waves * 16 rows
    const int nb = n / rowsPerBlock;      // 16

    const size_t shmem = ((size_t)D * 32 + 16) * sizeof(float); // 128KB + reduce

    nce_main_kernel<<<nb, 256, shmem, stream>>>(hidden, tgt, W, noise, parts,
                                                n, D, V, K);
    nce_final_kernel<<<1, 32, 0, stream>>>(parts, out, nb, n, K);
}